// SlicingLayer_42580305773133
// MI455X (gfx1250) — compile-verified
//
#include <hip/hip_runtime.h>
#include <hip/hip_bf16.h>

namespace {

constexpr int N  = 4;
constexpr int C  = 12;
constexpr int D  = 8;
constexpr int Hg = 16;
constexpr int Wg = 16;
constexpr int H  = 1024;
constexpr int W  = 1024;

constexpr int THREADS       = 256;
constexpr int PX_PER_THREAD = 4;            // 256 * 4 = 1024 = one image row per block

constexpr int A_X       = Wg + 1;           // x slot 16 holds zeros => exact x-OOB mask
constexpr int A_ENTRIES = A_X * D * C;      // 17*8*12 = 1632 floats (6.4 KB)
constexpr int A_REAL    = Wg * D * C;       // 1536 computed entries

constexpr int PLANES    = C * D;            // 96 (c,z) planes, stride Hg*Wg in memory
constexpr int STAGE_FLT = PLANES * 2 * Wg;  // TDM tile: [plane][ypair][x] = 3072 floats

#if __has_builtin(__builtin_amdgcn_tensor_load_to_lds)
#define USE_TDM 1
typedef unsigned int u32x4 __attribute__((ext_vector_type(4)));
typedef int          i32x8 __attribute__((ext_vector_type(8)));
typedef int          i32x4 __attribute__((ext_vector_type(4)));
#else
#define USE_TDM 0
#endif

struct SMem {
    float  stage[STAGE_FLT];        // must be first => LDS byte offset 0 for the D#
    float4 a4[A_ENTRIES / 4];       // y-lerped table A[x][z][c]
};

__global__ __launch_bounds__(THREADS) void slice_kernel(
    const float* __restrict__ grid,    // [N][C][D][Hg][Wg]
    const float* __restrict__ guide,   // [N][1][H][W]
    float* __restrict__ out)           // [N][C][H][W]
{
    __shared__ SMem sm;
    float* sA = reinterpret_cast<float*>(sm.a4);

    const int row = blockIdx.x;            // n*H + h
    const int n   = row >> 10;
    const int h   = row & (H - 1);
    const int tid = threadIdx.x;

    const float* gRow = guide + (size_t)row * W;
    // warm the guidance row while we build the table (gfx1250 global_prefetch_b8)
    __builtin_prefetch(gRow + tid * PX_PER_THREAD, 0, 3);

    // ---------------- row-uniform y-lerp setup ----------------
    const float iy  = (float)h * (15.0f / 1023.0f);
    const float y0f = floorf(iy);
    const float fy  = iy - y0f;
    const int   y0  = (int)y0f;            // always in [0, Hg-1]
    const int   y1  = y0 + 1;
    const float wy0 = 1.0f - fy;
    const float wy1 = (y1 < Hg) ? fy : 0.0f;

    const float* gbase = grid + (size_t)n * (PLANES * Hg * Wg);

#if USE_TDM
    // ---- Tensor Data Mover: pull the (y0, y0+1) row pair of every (c,z)
    // plane into LDS as one 3-D tile: x=16/stride 1, y=2/stride 16,
    // plane=96/stride 256 (4-byte elements, 12 KB). Rows past the tensor edge
    // (y0==15) read back as zeros, which is exactly the reference's y mask.
    if (tid < 32) {                        // wave 0 only: TDM ignores EXEC
        const unsigned long long ga =
            (unsigned long long)(uintptr_t)(gbase + y0 * Wg);   // tile start
        const u32x4 g0 = {
            1u,                            // count=1, user descriptor
            0u,                            // lds_addr: sm.stage at LDS offset 0
            (unsigned)ga,                  // global_addr[31:0]
            (unsigned)(ga >> 32) | 0x80000000u  // global_addr[56:32] | type=2
        };
        const i32x8 g1 = {
            2 << 16,                       // data_size = 4-byte elements
            Wg << 16,                      // tensor_dim0 = 16 (low half)
            (Hg - y0) << 16,               // tensor_dim1 = rows left below edge
            Wg << 16,                      // tile_dim0 = 16
            (PLANES << 16) | 2,            // tile_dim1 = 2, tile_dim2 = 96
            Wg,                            // tensor_dim0_stride = 16
            (Hg * Wg) << 16,               // tensor_dim1_stride = 256 (plane)
            0
        };
        const i32x4 g2 = { PLANES, 0, 0, 0 };   // tensor_dim2 = 96; dim3 unused
        const i32x4 g3 = { 0, 0, 0, 0 };
        const i32x8 g4 = { 0, 0, 0, 0, 0, 0, 0, 0 };  // clang-23 6-arg form: zero-filled
        __builtin_amdgcn_tensor_load_to_lds(g0, g1, g2, g3, g4, 0);
        __builtin_amdgcn_s_wait_tensorcnt(0);
    }
    __syncthreads();
#endif

    // ---------------- phase 1: fold the y-lerp into the LDS table ----------------
    for (int e = tid; e < A_ENTRIES; e += THREADS) {
        float a = 0.0f;                    // x==16 slots stay zero
        if (e < A_REAL) {
            const int c = e % C;
            const int s = e / C;
            const int z = s % D;
            const int x = s / D;
#if USE_TDM
            const int p  = c * D + z;      // plane index in the staged tile
            const float v0 = sm.stage[p * (2 * Wg) + x];
            const float v1 = sm.stage[p * (2 * Wg) + Wg + x];
#else
            const int   yc1 = (y1 < Hg) ? y1 : (Hg - 1);
            const int   cz  = (c * D + z) * (Hg * Wg);
            const float v0  = gbase[cz + y0  * Wg + x];
            const float v1  = gbase[cz + yc1 * Wg + x];
#endif
            a = fmaf(wy1, v1, wy0 * v0);
        }
        sA[e] = a;                         // layout index == e == (x*D+z)*C + c
    }
    __syncthreads();

    // ---------------- phase 2: 4 pixels per thread ----------------
    const int wb = tid * PX_PER_THREAD;
    const float4 g4v = *reinterpret_cast<const float4*>(gRow + wb);
    const float gv[PX_PER_THREAD] = {g4v.x, g4v.y, g4v.z, g4v.w};

    float acc[C * PX_PER_THREAD];          // acc[c*4 + px]

#pragma unroll
    for (int px = 0; px < PX_PER_THREAD; ++px) {
        const int   w   = wb + px;
        const float ix  = (float)w * (15.0f / 1023.0f);
        const float x0f = floorf(ix);
        const float fx  = ix - x0f;
        const int   x0  = (int)x0f;        // in [0, 15]
        const float wx0 = 1.0f - fx;
        const float wx1 = fx;              // A[16]==0 encodes the x OOB mask exactly

        const float iz  = (gv[px] + 1.0f) * (0.5f * (float)(D - 1));
        const float z0f = floorf(iz);
        const float fz  = iz - z0f;
        const int   z0  = (int)z0f;
        const int   z1  = z0 + 1;
        const float wz0 = (z0 >= 0 && z0 < D) ? (1.0f - fz) : 0.0f;
        const float wz1 = (z1 >= 0 && z1 < D) ? fz : 0.0f;
        const int   zc0 = min(max(z0, 0), D - 1);
        const int   zc1 = min(max(z1, 0), D - 1);

        const float4* A00 = sm.a4 + ((x0)     * D + zc0) * (C / 4);
        const float4* A10 = sm.a4 + ((x0 + 1) * D + zc0) * (C / 4);
        const float4* A01 = sm.a4 + ((x0)     * D + zc1) * (C / 4);
        const float4* A11 = sm.a4 + ((x0 + 1) * D + zc1) * (C / 4);

#pragma unroll
        for (int q = 0; q < C / 4; ++q) {
            const float4 a00 = A00[q];
            const float4 a10 = A10[q];
            const float4 a01 = A01[q];
            const float4 a11 = A11[q];

#define COMB(comp, cc)                                                        \
            {                                                                 \
                const float r0 = fmaf(wx1, a10.comp, wx0 * a00.comp);         \
                const float r1 = fmaf(wx1, a11.comp, wx0 * a01.comp);         \
                acc[(cc) * PX_PER_THREAD + px] = fmaf(wz1, r1, wz0 * r0);     \
            }
            COMB(x, q * 4 + 0)
            COMB(y, q * 4 + 1)
            COMB(z, q * 4 + 2)
            COMB(w, q * 4 + 3)
#undef COMB
        }
    }

    // ---------------- stores: 12 coalesced b128 streams ----------------
#pragma unroll
    for (int c = 0; c < C; ++c) {
        const float4 v = make_float4(acc[c * 4 + 0], acc[c * 4 + 1],
                                     acc[c * 4 + 2], acc[c * 4 + 3]);
        float* dst = out + (((size_t)(n * C + c) * H + h) * W + wb);
        *reinterpret_cast<float4*>(dst) = v;
    }
}

} // anonymous namespace

extern "C" void kernel_launch(void* const* d_in, const int* in_sizes, int n_in,
                              void* d_out, int out_size, void* d_ws, size_t ws_size,
                              hipStream_t stream) {
    (void)in_sizes; (void)n_in; (void)out_size; (void)d_ws; (void)ws_size;
    const float* grid  = (const float*)d_in[0];   // (4,12,8,16,16) fp32
    const float* guide = (const float*)d_in[1];   // (4,1,1024,1024) fp32
    float*       out   = (float*)d_out;           // (4,12,1024,1024) fp32

    slice_kernel<<<dim3(N * H), dim3(THREADS), 0, stream>>>(grid, guide, out);
}